// Laplacian_58763742544582
// MI455X (gfx1250) — compile-verified
//
#include <hip/hip_runtime.h>
#include <hip/hip_bf16.h>
#include <math.h>

// ---------------------------------------------------------------------------
// Exact Laplacian of 3-layer tanh MLP, closed form.
//   D=8192 input dim, H=128 hidden.  Outputs: [f0 x D | grad x D | lap]
// Heavy op: G = W1^T W1 (128x128, K=8192) on V_WMMA_F32_16X16X4_F32.
// LDS tile is pair-interleaved so each WMMA fragment is a single ds_load_b64.
// ---------------------------------------------------------------------------

#define DD 8192
#define HH 128
#define ROWS_PER_CHUNK 128
#define NCHUNK (DD / ROWS_PER_CHUNK) /* 64 */

typedef __attribute__((ext_vector_type(2))) float v2f;
typedef __attribute__((ext_vector_type(8))) float v8f;

// ---------------------------------------------------------------------------
// K1: per-chunk partial Gram (WMMA f32) + partial z1.
// grid = NCHUNK blocks, block = 256 threads (8 wave32).
// LDS layout: element (i,j) of the chunk lives at
//   lds[((i>>1)*HH + j)*2 + (i&1)]
// so rows (2r, 2r+1) of column j are adjacent -> one b64 per WMMA operand.
// ---------------------------------------------------------------------------
__global__ void __launch_bounds__(256)
k1_gram_z1(const float* __restrict__ W1, const float* __restrict__ x,
           float* __restrict__ pG, float* __restrict__ pZ) {
  __shared__ float lds[ROWS_PER_CHUNK * HH];  // 64 KB, pair-interleaved
  __shared__ float ztmp[2 * HH];

  const int blk = blockIdx.x;
  const int tid = threadIdx.x;
  const int i0  = blk * ROWS_PER_CHUNK;

  // Stage chunk: thread handles (row-pair rp, col j); global reads coalesced
  // over j, LDS write is one contiguous 8-byte pair.
  for (int idx = tid; idx < (ROWS_PER_CHUNK / 2) * HH; idx += 256) {
    const int rp = idx >> 7;       // 0..63  row pair
    const int j  = idx & (HH - 1); // 0..127 column
    const size_t g = (size_t)(i0 + rp * 2) * HH + j;
    v2f p;
    p.x = W1[g];
    p.y = W1[g + HH];
    *(v2f*)&lds[(size_t)idx * 2] = p;
  }
  __syncthreads();

  // Partial z1_j = sum_{i in chunk} x_i * W1[i,j]
  {
    const int j    = tid & (HH - 1);
    const int half = tid >> 7;  // 0 or 1
    float acc = 0.f;
    const int ib = half * 64;
    for (int i = ib; i < ib + 64; ++i)
      acc += x[i0 + i] * lds[((i >> 1) * HH + j) * 2 + (i & 1)];
    ztmp[half * HH + j] = acc;
  }
  __syncthreads();
  if (tid < HH) pZ[blk * HH + tid] = ztmp[tid] + ztmp[HH + tid];

  // Partial Gram: G += Wc^T Wc.  Wave w owns output tile-row w (16 rows),
  // sweeps 8 tiles of 16 cols, K-loop over 128 chunk rows in steps of 4.
  const int wave = tid >> 5;
  const int lane = tid & 31;
  const int m    = lane & 15;
  const int ks   = (lane < 16) ? 0 : 2;  // K offset for this half-wave
  const int j0   = wave * 16;

  v8f acc[8];
#pragma unroll
  for (int t = 0; t < 8; ++t) acc[t] = v8f{};

  for (int k0 = 0; k0 < ROWS_PER_CHUNK; k0 += 4) {
    const int rp = (k0 + ks) >> 1;  // aligned row pair (k0+ks even)
    // A[m,k] = Wc[k0+k, j0+m]  -> contiguous pair, single ds_load_b64
    const v2f a = *(const v2f*)&lds[((size_t)rp * HH + (j0 + m)) * 2];
#pragma unroll
    for (int t = 0; t < 8; ++t) {
      // B[k,n] = Wc[k0+k, t*16+n] -> contiguous pair
      const v2f b = *(const v2f*)&lds[((size_t)rp * HH + (t * 16 + m)) * 2];
      acc[t] = __builtin_amdgcn_wmma_f32_16x16x4_f32(
          false, a, false, b, (short)0, acc[t], false, false);
    }
  }

  // Store partial tiles. C layout: VGPR r -> lanes0-15 row j0+r, col=lane;
  // lanes16-31 row j0+8+r, col=lane-16.
  float* g = pG + (size_t)blk * (HH * HH);
  const int rowb = j0 + ((lane < 16) ? 0 : 8);
#pragma unroll
  for (int t = 0; t < 8; ++t) {
    const int col = t * 16 + m;
#pragma unroll
    for (int r = 0; r < 8; ++r) g[(rowb + r) * HH + col] = acc[t][r];
  }
}

// ---------------------------------------------------------------------------
// K2: deterministic reduction of partial Grams.  grid = 64 x 256 threads.
// ---------------------------------------------------------------------------
__global__ void __launch_bounds__(256)
k2_reduce_g(const float* __restrict__ pG, float* __restrict__ G) {
  const int idx = blockIdx.x * 256 + threadIdx.x;  // 0..16383
  float s = 0.f;
  for (int b = 0; b < NCHUNK; ++b) s += pG[(size_t)b * (HH * HH) + idx];
  G[idx] = s;
}

// ---------------------------------------------------------------------------
// K2b: z1 reduction + layer-1 activations.  1 block x 128 threads.
// ---------------------------------------------------------------------------
__global__ void __launch_bounds__(HH)
k2b_act(const float* __restrict__ pZ, const float* __restrict__ b1,
        float* __restrict__ h1, float* __restrict__ t1,
        float* __restrict__ t1dd) {
  const int j = threadIdx.x;
  float s = b1[j];
  for (int b = 0; b < NCHUNK; ++b) s += pZ[b * HH + j];
  const float h = tanhf(s);
  const float t = 1.f - h * h;
  h1[j] = h;
  t1[j] = t;
  t1dd[j] = -2.f * h * t;  // tanh''
}

// ---------------------------------------------------------------------------
// K3: head math — z2/h2, g1 = t1 .* (W2 g2), both Laplacian terms, f0.
// 1 block x 128 threads.  Writes out[2D] = Laplacian, sc[0] = f0.
// ---------------------------------------------------------------------------
__global__ void __launch_bounds__(HH)
k3_head(const float* __restrict__ W2, const float* __restrict__ b2,
        const float* __restrict__ W3, const float* __restrict__ b3,
        const float* __restrict__ G, const float* __restrict__ h1,
        const float* __restrict__ t1, const float* __restrict__ t1dd,
        float* __restrict__ g1, float* __restrict__ sc,
        float* __restrict__ out) {
  __shared__ float s_h1[HH], s_t1[HH], s_g2[HH], s_red[HH];
  __shared__ float s_m[HH * HH];  // M[j][k] = t1[j] * W2[j,k]  (64 KB)

  const int k = threadIdx.x;
  s_h1[k] = h1[k];
  s_t1[k] = t1[k];
  __syncthreads();

  // z2_k = b2_k + sum_j W2[j,k] h1_j   (coalesced over k)
  float z2 = b2[k];
  for (int j = 0; j < HH; ++j) z2 += W2[j * HH + k] * s_h1[j];
  const float h2   = tanhf(z2);
  const float t2   = 1.f - h2 * h2;
  const float t2dd = -2.f * h2 * t2;
  const float w3k  = W3[k];
  s_g2[k] = t2 * w3k;
  for (int j = 0; j < HH; ++j) s_m[j * HH + k] = s_t1[j] * W2[j * HH + k];
  __syncthreads();

  // u_j = sum_k W2[j,k] g2_k ;  g1_j = t1_j u_j ;  term2_j = t1''_j u_j G_jj
  const int j = k;
  float u = 0.f;
  for (int kk = 0; kk < HH; ++kk) u += W2[j * HH + kk] * s_g2[kk];
  g1[j] = s_t1[j] * u;
  const float term2 = t1dd[j] * u * G[j * HH + j];

  // C_kk = m_k^T G m_k  (G loads broadcast; s_m conflict-free over k)
  float ckk = 0.f;
  for (int jj = 0; jj < HH; ++jj) {
    float v = 0.f;
    for (int j2 = 0; j2 < HH; ++j2) v += G[jj * HH + j2] * s_m[j2 * HH + k];
    ckk += s_m[jj * HH + k] * v;
  }
  const float term1 = w3k * t2dd * ckk;

  // reduce lap = sum_k(term1) + sum_j(term2);  f0 = sum_k W3_k h2_k + b3
  s_red[k] = term1 + term2;
  __syncthreads();
  for (int s = 64; s > 0; s >>= 1) {
    if (k < s) s_red[k] += s_red[k + s];
    __syncthreads();
  }
  const float lap = s_red[0];
  __syncthreads();
  s_red[k] = w3k * h2;
  __syncthreads();
  for (int s = 64; s > 0; s >>= 1) {
    if (k < s) s_red[k] += s_red[k + s];
    __syncthreads();
  }
  if (k == 0) {
    sc[0] = s_red[0] + b3[0];  // f0
    out[2 * DD] = lap;         // F2.sum
  }
}

// ---------------------------------------------------------------------------
// K4: grad = W1 * g1, broadcast f0.  grid = 32 x 256.
// ---------------------------------------------------------------------------
__global__ void __launch_bounds__(256)
k4_grad_out(const float* __restrict__ W1, const float* __restrict__ g1,
            const float* __restrict__ sc, float* __restrict__ out) {
  __shared__ float s_g1[HH];
  const int tid = threadIdx.x;
  if (tid < HH) s_g1[tid] = g1[tid];
  __syncthreads();
  const int i = blockIdx.x * 256 + tid;
  const float4* row = (const float4*)(W1 + (size_t)i * HH);
  float acc = 0.f;
#pragma unroll
  for (int q = 0; q < HH / 4; ++q) {
    const float4 w = row[q];
    acc += w.x * s_g1[4 * q] + w.y * s_g1[4 * q + 1] + w.z * s_g1[4 * q + 2] +
           w.w * s_g1[4 * q + 3];
  }
  out[i]      = sc[0];  // F0 = f(x) replicated
  out[DD + i] = acc;    // F1 = gradient
}

// ---------------------------------------------------------------------------
// Launch.  Inputs (dict order): x, W1, b1, W2, b2, W3, b3.
// ---------------------------------------------------------------------------
extern "C" void kernel_launch(void* const* d_in, const int* in_sizes, int n_in,
                              void* d_out, int out_size, void* d_ws,
                              size_t ws_size, hipStream_t stream) {
  const float* x  = (const float*)d_in[0];
  const float* W1 = (const float*)d_in[1];
  const float* b1 = (const float*)d_in[2];
  const float* W2 = (const float*)d_in[3];
  const float* b2 = (const float*)d_in[4];
  const float* W3 = (const float*)d_in[5];
  const float* b3 = (const float*)d_in[6];
  float* out = (float*)d_out;

  float* w = (float*)d_ws;
  float* pG   = w;                                   // 64*128*128
  float* pZ   = pG + (size_t)NCHUNK * HH * HH;       // 64*128
  float* G    = pZ + (size_t)NCHUNK * HH;            // 128*128
  float* h1   = G + HH * HH;
  float* t1   = h1 + HH;
  float* t1dd = t1 + HH;
  float* g1   = t1dd + HH;
  float* sc   = g1 + HH;                             // [f0]

  k1_gram_z1<<<NCHUNK, 256, 0, stream>>>(W1, x, pG, pZ);
  k2_reduce_g<<<(HH * HH) / 256, 256, 0, stream>>>(pG, G);
  k2b_act<<<1, HH, 0, stream>>>(pZ, b1, h1, t1, t1dd);
  k3_head<<<1, HH, 0, stream>>>(W2, b2, W3, b3, G, h1, t1, t1dd, g1, sc, out);
  k4_grad_out<<<DD / 256, 256, 0, stream>>>(W1, g1, sc, out);
}